// GraphEncoder_87333864997319
// MI455X (gfx1250) — compile-verified
//
#include <hip/hip_runtime.h>

// ---------------------------------------------------------------------------
// CDNA5 (gfx1250) GCN-VAE encoder.
// GEMMs: native fp32 WMMA (V_WMMA_F32_16X16X4_F32), wave32.
//   - one wave computes a 16x64 output strip (4 accumulator tiles)
//   - 8 waves/block share a 64(K) x 64(N) B-chunk staged in LDS
//     (coalesced b128 global loads -> ds_store_b128; fragments via ds_load)
//   - LDS row stride 72 (== 8 mod 64) puts the two lane-halves on disjoint
//     bank sets (offset 16) -> conflict-free B fragment reads.
// Edge aggregation: global_atomic_add_f32; all intermediates are L2-resident
// (192 MB L2 >> 128 MB working set), so the scatter is the bandwidth bound,
// not the GEMMs -> fp32 precision is free.
// ---------------------------------------------------------------------------

typedef __attribute__((ext_vector_type(2))) float v2f;
typedef __attribute__((ext_vector_type(8))) float v8f;

// ---- small utility kernels -------------------------------------------------

__global__ void k_fill(float* __restrict__ p, float v, int n) {
    int i = blockIdx.x * blockDim.x + threadIdx.x;
    if (i < n) p[i] = v;
}

__global__ void k_zero(float* __restrict__ p, long total) {
    long i = (long)blockIdx.x * blockDim.x + threadIdx.x;
    long stride = (long)gridDim.x * blockDim.x;
    for (; i < total; i += stride) p[i] = 0.0f;
}

// degree accumulation: deg pre-filled with 1.0 (self loop), +1 per in-edge
__global__ void k_deg(const int* __restrict__ dst, float* __restrict__ deg, int E) {
    int i = blockIdx.x * blockDim.x + threadIdx.x;
    if (i < E) atomicAdd(&deg[dst[i]], 1.0f);
}

// deg -> d^{-1/2} in place (deg >= 1 always, thanks to self loop)
__global__ void k_rsqrt(float* __restrict__ deg, int n) {
    int i = blockIdx.x * blockDim.x + threadIdx.x;
    if (i < n) deg[i] = rsqrtf(deg[i]);
}

// ---- edge scatter: agg[dst,c] += dinv[src]*dinv[dst] * h[src,c] -----------
__global__ void k_scatter(const float* __restrict__ h,
                          const int* __restrict__ src,
                          const int* __restrict__ dst,
                          const float* __restrict__ dinv,
                          float* __restrict__ agg,
                          long total, int shift, int cmask) {
    long i = (long)blockIdx.x * blockDim.x + threadIdx.x;
    long stride = (long)gridDim.x * blockDim.x;
    for (; i < total; i += stride) {
        int e = (int)(i >> shift);
        int c = (int)(i & cmask);
        int s = src[e];
        int d = dst[e];
        float nrm = dinv[s] * dinv[d];
        atomicAdd(&agg[((long)d << shift) + c], h[((long)s << shift) + c] * nrm);
    }
}

// ---- finalize: agg = relu(agg + dinv[node]^2 * h[node] + bias) ------------
__global__ void k_finalize(float* __restrict__ agg,
                           const float* __restrict__ h,
                           const float* __restrict__ dinv,
                           const float* __restrict__ bias,
                           long total, int shift, int cmask) {
    long i = (long)blockIdx.x * blockDim.x + threadIdx.x;
    long stride = (long)gridDim.x * blockDim.x;
    for (; i < total; i += stride) {
        int node = (int)(i >> shift);
        int c = (int)(i & cmask);
        float di = dinv[node];
        float v = agg[i] + di * di * h[i] + bias[c];
        agg[i] = v > 0.0f ? v : 0.0f;
    }
}

// ---- fp32 WMMA GEMM: C[M,N] = A[M,K] @ B[K,N] (+bias) ---------------------
// Requires M%16==0, K%64==0, N%64==0  (M=50000=3125*16, K,N in {128,256}).
// Block: 256 threads = 8 waves = 8 consecutive M-tiles x one 64-col N-strip.
#define KB 64            // K-chunk staged in LDS per iteration
#define LDS_STRIDE 72    // 72 = 8 (mod 64): lane-halves hit disjoint banks

#define WMMA_STEP(ACC, OFF)                                                    \
    do {                                                                       \
        v2f b;                                                                 \
        b.x = bro[OFF];                                                        \
        b.y = br1[OFF];                                                        \
        ACC = __builtin_amdgcn_wmma_f32_16x16x4_f32(                           \
            false, a, false, b, (short)0, ACC, false, false);                  \
    } while (0)

__global__ __launch_bounds__(256)
void k_gemm_wmma_f32(const float* __restrict__ A,
                     const float* __restrict__ B,
                     const float* __restrict__ bias,
                     float* __restrict__ C,
                     int M, int K, int N, int addBias) {
    __shared__ float sB[KB * LDS_STRIDE];   // 18 KB

    const int tid  = threadIdx.x;
    const int lane = tid & 31;
    const int wv   = tid >> 5;              // wave in block: 0..7
    const int stripsN = N >> 6;             // 64-col strips
    const int tilesM  = M >> 4;
    const int bm = blockIdx.x / stripsN;
    const int sn = blockIdx.x - bm * stripsN;

    const int tm    = bm * 8 + wv;
    const int live  = tm < tilesM;          // wave-uniform
    const int tmEff = live ? tm : (tilesM - 1);   // clamp: keep loads legal,
                                                  // keep EXEC full for WMMA
    const int m0 = tmEff << 4;
    const int n0 = sn << 6;

    const int half = lane >> 4;             // 0: K{k,k+1}, 1: K{k+2,k+3}
    const int l    = lane & 15;

    const float* arow = A + (long)(m0 + l) * K + half * 2;

    v8f acc0 = {}, acc1 = {}, acc2 = {}, acc3 = {};

    for (int kb = 0; kb < K; kb += KB) {
        __syncthreads();   // previous chunk fully consumed before overwrite
        // cooperative stage: B[kb..kb+64) x [n0..n0+64) -> LDS, float4 lanes
        for (int i = tid; i < (KB * 64) / 4; i += 256) {
            int r = (i * 4) >> 6;           // row in chunk
            int c = (i * 4) & 63;           // col in strip (multiple of 4)
            float4 v = *(const float4*)(B + (long)(kb + r) * N + n0 + c);
            *(float4*)(&sB[r * LDS_STRIDE + c]) = v;   // ds_store_b128
        }
        __syncthreads();

        for (int k = 0; k < KB; k += 4) {
            v2f a;                           // A fragment, reused by 4 WMMAs
            a.x = arow[kb + k];
            a.y = arow[kb + k + 1];
            const float* bro = &sB[(k + half * 2) * LDS_STRIDE + l];
            const float* br1 = bro + LDS_STRIDE;
            WMMA_STEP(acc0, 0);
            WMMA_STEP(acc1, 16);
            WMMA_STEP(acc2, 32);
            WMMA_STEP(acc3, 48);
        }
    }

    if (live) {
        float bv0 = 0.f, bv1 = 0.f, bv2 = 0.f, bv3 = 0.f;
        if (addBias) {
            bv0 = bias[n0 + l];
            bv1 = bias[n0 + 16 + l];
            bv2 = bias[n0 + 32 + l];
            bv3 = bias[n0 + 48 + l];
        }
        // C/D layout: VGPR r -> row m0 + 8*half + r, col n0 + 16*t + l
        const long base = (long)(m0 + half * 8) * N + n0 + l;
#pragma unroll
        for (int r = 0; r < 8; ++r) {
            long ro = base + (long)r * N;
            C[ro +  0] = acc0[r] + bv0;
            C[ro + 16] = acc1[r] + bv1;
            C[ro + 32] = acc2[r] + bv2;
            C[ro + 48] = acc3[r] + bv3;
        }
    }
}

// ---------------------------------------------------------------------------

static inline int ilog2(int v) { return 31 - __builtin_clz((unsigned)v); }

static inline int gemm_blocks(int M, int N) {
    return (((M >> 4) + 7) / 8) * (N >> 6);
}

extern "C" void kernel_launch(void* const* d_in, const int* in_sizes, int n_in,
                              void* d_out, int out_size, void* d_ws, size_t ws_size,
                              hipStream_t stream) {
    const float* x   = (const float*)d_in[0];
    const int*   ei  = (const int*)  d_in[1];   // (2,E): src row then dst row
    const float* W1  = (const float*)d_in[2];
    const float* b1  = (const float*)d_in[3];
    const float* W2  = (const float*)d_in[4];
    const float* b2  = (const float*)d_in[5];
    const float* muW = (const float*)d_in[6];
    const float* mub = (const float*)d_in[7];
    const float* lvW = (const float*)d_in[8];
    const float* lvb = (const float*)d_in[9];

    const int IN  = 128;
    const int HID = in_sizes[3];            // 256
    const int LAT = in_sizes[5];            // 128
    const int n   = in_sizes[0] / IN;       // 50000
    const int E   = in_sizes[1] / 2;        // 800000

    const int* src = ei;
    const int* dst = ei + E;

    // workspace layout (floats): dinv | buf1 (n*HID) | buf2 (n*HID)
    float* ws   = (float*)d_ws;
    float* dinv = ws;
    float* buf1 = ws + (((long)n + 63) & ~63L);
    float* buf2 = buf1 + (long)n * HID;

    const int T = 256;
    const int shiftH = ilog2(HID), maskH = HID - 1;
    const int shiftL = ilog2(LAT), maskL = LAT - 1;

    // 1) symmetric normalization coefficients: dinv = rsqrt(1 + in-degree)
    k_fill <<<(n + T - 1) / T, T, 0, stream>>>(dinv, 1.0f, n);
    k_deg  <<<(E + T - 1) / T, T, 0, stream>>>(dst, dinv, E);
    k_rsqrt<<<(n + T - 1) / T, T, 0, stream>>>(dinv, n);

    // 2) layer 1: h1 = x @ W1  (buf1)
    k_gemm_wmma_f32<<<gemm_blocks(n, HID), T, 0, stream>>>(
        x, W1, nullptr, buf1, n, IN, HID, 0);
    {
        long totA = (long)n * HID;
        long totE = (long)E << shiftH;
        k_zero<<<(int)((totA + T - 1) / T), T, 0, stream>>>(buf2, totA);
        k_scatter<<<(int)((totE + T - 1) / T), T, 0, stream>>>(
            buf1, src, dst, dinv, buf2, totE, shiftH, maskH);
        k_finalize<<<(int)((totA + T - 1) / T), T, 0, stream>>>(
            buf2, buf1, dinv, b1, totA, shiftH, maskH);
    }

    // 3) layer 2: h2 = relu1 @ W2  (buf1, reusing dead h1)
    k_gemm_wmma_f32<<<gemm_blocks(n, LAT), T, 0, stream>>>(
        buf2, W2, nullptr, buf1, n, HID, LAT, 0);
    {
        long totA = (long)n * LAT;
        long totE = (long)E << shiftL;
        k_zero<<<(int)((totA + T - 1) / T), T, 0, stream>>>(buf2, totA);
        k_scatter<<<(int)((totE + T - 1) / T), T, 0, stream>>>(
            buf1, src, dst, dinv, buf2, totE, shiftL, maskL);
        k_finalize<<<(int)((totA + T - 1) / T), T, 0, stream>>>(
            buf2, buf1, dinv, b2, totA, shiftL, maskL);
    }

    // 4) heads: mu = z @ muW + mub ; log_var = z @ lvW + lvb  (into d_out)
    {
        float* mu = (float*)d_out;
        float* lv = mu + (long)n * LAT;
        k_gemm_wmma_f32<<<gemm_blocks(n, LAT), T, 0, stream>>>(
            buf2, muW, mub, mu, n, LAT, LAT, 1);
        k_gemm_wmma_f32<<<gemm_blocks(n, LAT), T, 0, stream>>>(
            buf2, lvW, lvb, lv, n, LAT, LAT, 1);
    }
}